// AutoCorrelationLayer_18494129177301
// MI455X (gfx1250) — compile-verified
//
#include <hip/hip_runtime.h>
#include <hip/hip_bf16.h>
#include <math.h>

// ---------------- problem constants (match reference) ----------------
#define B_    16
#define L_    3072
#define D_    512
#define H_    8
#define DH_   64
#define TOPK  40
#define CH_   (B_ * D_)          // 8192 channels (b, h, dh)
#define M_    (B_ * L_)          // 49152 GEMM rows
#define LAGS_PER_THR 12          // L_ / 256
#define PADW  256                // circular-wrap padding for correlation LDS

typedef __attribute__((ext_vector_type(2))) float v2f;
typedef __attribute__((ext_vector_type(8))) float v8f;

// =====================================================================
// 1) fp32 WMMA GEMM:  Y[M,N] = X[M,K] @ W[N,K]^T + bias[N]
//    (torch Linear semantics). 256 threads = 8 waves, each wave owns a
//    16x16 C tile; block tile = 64 rows x 32 cols. K consumed 4/step via
//    V_WMMA_F32_16X16X4_F32.
// =====================================================================
__global__ __launch_bounds__(256)
void wmma_gemm_xWT_bias(const float* __restrict__ X,
                        const float* __restrict__ W,
                        const float* __restrict__ bias,
                        float* __restrict__ Y,
                        int M, int N, int K) {
    const int tid    = threadIdx.x;
    const int waveId = tid >> 5;
    const int lane   = tid & 31;
    const int l16    = lane & 15;
    const int lh     = lane >> 4;              // 0: K pair {0,1}, 1: {2,3}

    const int rowBase = blockIdx.y * 64 + (waveId >> 1) * 16;
    const int colBase = blockIdx.x * 32 + (waveId & 1) * 16;

    const int aRow = rowBase + l16;            // A: lane holds M=l16, K pair by lh
    const int bRow = colBase + l16;            // B[k][n] = W[n][k] -> lane holds N=l16

    const float* __restrict__ pa = X + (size_t)aRow * K + lh * 2;
    const float* __restrict__ pb = W + (size_t)bRow * K + lh * 2;

    v8f acc = {};
#pragma unroll 4
    for (int kk = 0; kk < K; kk += 4) {
        v2f a = *reinterpret_cast<const v2f*>(pa + kk);
        v2f b = *reinterpret_cast<const v2f*>(pb + kk);
        acc = __builtin_amdgcn_wmma_f32_16x16x4_f32(
                  false, a, false, b, (short)0, acc, false, false);
    }

    const int   n    = colBase + l16;
    const float bval = bias[n];
#pragma unroll
    for (int v = 0; v < 8; ++v) {
        int m = rowBase + v + lh * 8;          // C layout: VGPR v -> M=v (+8 for hi half)
        Y[(size_t)m * N + n] = acc[v] + bval;
    }
}

// =====================================================================
// 2) [B,L,D] -> [B,D,L] tiled transpose (channel-major for correlation)
// =====================================================================
__global__ __launch_bounds__(256)
void transpose_BLD_to_BDL(const float* __restrict__ in, float* __restrict__ out) {
    __shared__ float tile[32][33];
    const int b  = blockIdx.z;
    const int l0 = blockIdx.y * 32;
    const int d0 = blockIdx.x * 32;
    const int tx = threadIdx.x;                // 0..31
    const int ty = threadIdx.y;                // 0..7
#pragma unroll
    for (int i = 0; i < 32; i += 8)
        tile[ty + i][tx] = in[(size_t)b * L_ * D_ + (size_t)(l0 + ty + i) * D_ + d0 + tx];
    __syncthreads();
#pragma unroll
    for (int i = 0; i < 32; i += 8)
        out[(size_t)b * D_ * L_ + (size_t)(d0 + ty + i) * L_ + l0 + tx] = tile[tx][ty + i];
}

// =====================================================================
// 3) Circular autocorrelation per channel, on the MATRIX pipe:
//    AC[ch, lag] = sum_t Qt[ch, (lag+t)%L] * Kt[ch, t]
//    (== irfft(rfft(Q) * conj(rfft(K)))).
//
//    Hankel-tile WMMA: A[m][k] = Q[tau0+m+t0+k], B[k][n] = K[t0+k-16n]
//    => D[m][n] accumulates lag (tau0 + m + 16n) from t' = t0+k-16n.
//    Sweeping t0 over 0..L step 4 covers every t' residue exactly once
//    (circular via +-PADW wrapped LDS copies). One wave's v8f accumulator
//    covers the 256 lags {m+16n}; 12 waves (384 threads) = one channel.
// =====================================================================
__global__ __launch_bounds__(384)
void autocorr_wmma(const float* __restrict__ Qt,
                   const float* __restrict__ Kt,
                   float* __restrict__ AC) {
    __shared__ float Qs[2 * L_ + PADW];   // Qs[PADW + j] = Q[j mod L], j in [-PADW, 2L)
    __shared__ float Ks[L_ + PADW];       // Ks[PADW + j] = K[j mod L], j in [-PADW, L)
    const int ch  = blockIdx.x;
    const int tid = threadIdx.x;
    const size_t base = (size_t)ch * L_;

    for (int i = tid; i < L_; i += 384) {
        float qv = Qt[base + i];
        Qs[PADW + i]      = qv;
        Qs[PADW + i + L_] = qv;
        Ks[PADW + i]      = Kt[base + i];
    }
    if (tid < PADW) {                     // wrap pads: index j in [-PADW, 0)
        Qs[tid] = Qt[base + (L_ - PADW) + tid];
        Ks[tid] = Kt[base + (L_ - PADW) + tid];
    }
    __syncthreads();

    const int wave = tid >> 5;
    const int lane = tid & 31;
    const int l16  = lane & 15;
    const int lh   = lane >> 4;
    const int tau0 = wave * 256;

    const float* __restrict__ pa = &Qs[PADW + tau0 + l16 + lh * 2];  // A[m][k]
    const float* __restrict__ pb = &Ks[PADW + lh * 2 - 16 * l16];    // B[k][n]

    v8f acc = {};
#pragma unroll 8
    for (int t0 = 0; t0 < L_; t0 += 4) {
        v2f a = *reinterpret_cast<const v2f*>(pa + t0);
        v2f b = *reinterpret_cast<const v2f*>(pb + t0);
        acc = __builtin_amdgcn_wmma_f32_16x16x4_f32(
                  false, a, false, b, (short)0, acc, false, false);
    }

    // stage through LDS so the global write is fully coalesced
    __syncthreads();
#pragma unroll
    for (int v = 0; v < 8; ++v)
        Qs[tau0 + (v + 8 * lh) + 16 * l16] = acc[v];   // lag = tau0 + m + 16n
    __syncthreads();
    for (int i = tid; i < L_; i += 384)
        AC[base + i] = Qs[i];
}

// =====================================================================
// 4) Top-40 per channel (values sorted descending, ties -> lower index,
//    matching jax.lax.top_k). Iterative parallel argmax with masking.
// =====================================================================
__global__ __launch_bounds__(256)
void topk_channel(const float* __restrict__ AC,
                  float* __restrict__ TV, int* __restrict__ TI) {
    __shared__ float vals[L_];
    __shared__ float redV[256];
    __shared__ int   redI[256];
    const int ch  = blockIdx.x;
    const int tid = threadIdx.x;
    const size_t base = (size_t)ch * L_;

    for (int i = tid; i < L_; i += 256) vals[i] = AC[base + i];
    __syncthreads();

    for (int pass = 0; pass < TOPK; ++pass) {
        float bv = -3.402823e38f;
        int   bi = 0;
#pragma unroll
        for (int r = 0; r < LAGS_PER_THR; ++r) {
            int   i = tid + r * 256;
            float v = vals[i];
            if (v > bv) { bv = v; bi = i; }    // strict > keeps smallest index
        }
        redV[tid] = bv; redI[tid] = bi;
        __syncthreads();
        for (int s = 128; s > 0; s >>= 1) {
            if (tid < s) {
                float v2 = redV[tid + s]; int i2 = redI[tid + s];
                if (v2 > redV[tid] || (v2 == redV[tid] && i2 < redI[tid])) {
                    redV[tid] = v2; redI[tid] = i2;
                }
            }
            __syncthreads();
        }
        if (tid == 0) {
            TV[(size_t)ch * TOPK + pass] = redV[0];
            TI[(size_t)ch * TOPK + pass] = redI[0];
            vals[redI[0]] = -3.402823e38f;
        }
        __syncthreads();
    }
}

// =====================================================================
// 5) softmax over the k dimension, per channel
// =====================================================================
__global__ __launch_bounds__(256)
void softmax_k(const float* __restrict__ TV, float* __restrict__ WG) {
    const int ch = blockIdx.x * 256 + threadIdx.x;
    if (ch >= CH_) return;
    const float* v = TV + (size_t)ch * TOPK;
    float m = -3.402823e38f;
#pragma unroll
    for (int k = 0; k < TOPK; ++k) m = fmaxf(m, v[k]);
    float e[TOPK], s = 0.0f;
#pragma unroll
    for (int k = 0; k < TOPK; ++k) { e[k] = expf(v[k] - m); s += e[k]; }
    const float inv = 1.0f / s;
    float* w = WG + (size_t)ch * TOPK;
#pragma unroll
    for (int k = 0; k < TOPK; ++k) w[k] = e[k] * inv;
}

// =====================================================================
// 6) shifts[k] = trunc(mean over channels of TI[ch,k]); integer-exact sum
// =====================================================================
__global__ __launch_bounds__(256)
void mean_shift(const int* __restrict__ TI, int* __restrict__ SH) {
    __shared__ int red[256];
    const int k   = blockIdx.x;
    const int tid = threadIdx.x;
    int sum = 0;
    for (int ch = tid; ch < CH_; ch += 256) sum += TI[(size_t)ch * TOPK + k];
    red[tid] = sum;
    __syncthreads();
    for (int s = 128; s > 0; s >>= 1) {
        if (tid < s) red[tid] += red[tid + s];
        __syncthreads();
    }
    if (tid == 0) SH[k] = (int)(((double)red[0]) / (double)CH_);
}

// =====================================================================
// 7) AGG[b,t,:] = sum_k WG[ch,k] * V[b,(t+shift_k)%L,:]   (roll(-s))
// =====================================================================
__global__ __launch_bounds__(256)
void aggregate_rolls(const float* __restrict__ V,
                     const float* __restrict__ WG,
                     const int* __restrict__ SH,
                     float* __restrict__ AGG) {
    __shared__ int sh_s[TOPK];
    if (threadIdx.x < TOPK) sh_s[threadIdx.x] = SH[threadIdx.x];
    __syncthreads();

    const int gid  = blockIdx.x * 256 + threadIdx.x;   // over B*L*D (exact multiple)
    const int dcol = gid % D_;
    const int bt   = gid / D_;
    const int t    = bt % L_;
    const int b    = bt / L_;
    const float* __restrict__ w = WG + ((size_t)b * D_ + dcol) * TOPK;

    float acc = 0.0f;
#pragma unroll 8
    for (int k = 0; k < TOPK; ++k) {
        int tv = t + sh_s[k];
        if (tv >= L_) tv -= L_;
        acc += w[k] * V[((size_t)b * L_ + tv) * D_ + dcol];
    }
    AGG[gid] = acc;
}

// =====================================================================
// launcher
// =====================================================================
extern "C" void kernel_launch(void* const* d_in, const int* in_sizes, int n_in,
                              void* d_out, int out_size, void* d_ws, size_t ws_size,
                              hipStream_t stream) {
    const float* query = (const float*)d_in[0];
    const float* key   = (const float*)d_in[1];
    const float* value = (const float*)d_in[2];
    const float* Wq    = (const float*)d_in[3];
    const float* bq    = (const float*)d_in[4];
    const float* Wk    = (const float*)d_in[5];
    const float* bk    = (const float*)d_in[6];
    const float* Wv    = (const float*)d_in[7];
    const float* bv    = (const float*)d_in[8];
    const float* Wo    = (const float*)d_in[9];
    const float* bo    = (const float*)d_in[10];
    float* out = (float*)d_out;

    // workspace layout (floats); big buffers reused once their producer role ends
    const size_t SZ = (size_t)B_ * L_ * D_;            // 25,165,824 floats
    float* ws   = (float*)d_ws;
    float* Qbld = ws + 0 * SZ;                         // Q [B,L,D]  -> reused as AC
    float* Kbld = ws + 1 * SZ;                         // K [B,L,D]  -> reused as AGG
    float* Vb   = ws + 2 * SZ;                         // V [B,L,D]
    float* Qt   = ws + 3 * SZ;                         // Q [B,D,L]
    float* Kt   = ws + 4 * SZ;                         // K [B,D,L]
    float* TV   = ws + 5 * SZ;                         // [CH, TOPK]
    float* WG   = TV + (size_t)CH_ * TOPK;             // [CH, TOPK]
    int*   TI   = (int*)(WG + (size_t)CH_ * TOPK);     // [CH, TOPK]
    int*   SH   = TI + (size_t)CH_ * TOPK;             // [TOPK]

    // 1) projections (WMMA fp32)
    dim3 gGemm(D_ / 32, M_ / 64);
    wmma_gemm_xWT_bias<<<gGemm, 256, 0, stream>>>(query, Wq, bq, Qbld, M_, D_, D_);
    wmma_gemm_xWT_bias<<<gGemm, 256, 0, stream>>>(key,   Wk, bk, Kbld, M_, D_, D_);
    wmma_gemm_xWT_bias<<<gGemm, 256, 0, stream>>>(value, Wv, bv, Vb,   M_, D_, D_);

    // 2) channel-major transposes for Q, K
    dim3 gT(D_ / 32, L_ / 32, B_), bT(32, 8);
    transpose_BLD_to_BDL<<<gT, bT, 0, stream>>>(Qbld, Qt);
    transpose_BLD_to_BDL<<<gT, bT, 0, stream>>>(Kbld, Kt);

    // 3) circular autocorrelation on the matrix pipe (AC reuses Q buffer)
    float* AC = Qbld;
    autocorr_wmma<<<CH_, 384, 0, stream>>>(Qt, Kt, AC);

    // 4) top-40 per channel
    topk_channel<<<CH_, 256, 0, stream>>>(AC, TV, TI);

    // 5) softmax over k
    softmax_k<<<(CH_ + 255) / 256, 256, 0, stream>>>(TV, WG);

    // 6) integer-exact mean shifts
    mean_shift<<<TOPK, 256, 0, stream>>>(TI, SH);

    // 7) weighted rolled aggregation (AGG reuses the K buffer)
    float* AGG = Kbld;
    aggregate_rolls<<<(int)(SZ / 256), 256, 0, stream>>>(Vb, WG, SH, AGG);

    // 8) output projection (WMMA fp32) straight into d_out
    wmma_gemm_xWT_bias<<<gGemm, 256, 0, stream>>>(AGG, Wo, bo, out, M_, D_, D_);
}